// VQVAE3D_45243185496182
// MI455X (gfx1250) — compile-verified
//
#include <hip/hip_runtime.h>

typedef __attribute__((ext_vector_type(2))) float v2f;
typedef __attribute__((ext_vector_type(8))) float v8f;

#define C_DIM      64
#define K_CODES    1024
#define N_SPATIAL  131072      // 32*64*64
#define QSIZE      16777216    // 2*64*131072
#define TOK_PER_WAVE 16
#define WAVES_PER_BLOCK 8
#define TOK_PER_BLOCK 128
#define NBLOCKS    2048        // 262144 / 128
#define ESTR       68          // padded LDS row stride (64 + 4) to spread banks

// ---------------------------------------------------------------------------
// Main kernel: fused distance-GEMM (f32 WMMA) + argmin + gather + loss partials
// ---------------------------------------------------------------------------
__global__ void vq_main(const float* __restrict__ x,
                        const float* __restrict__ emb,
                        float* __restrict__ outQ,
                        float* __restrict__ outIdx,
                        float* __restrict__ wsPartial)
{
    extern __shared__ float smem[];
    float* ldsE    = smem;                         // K_CODES * ESTR floats
    float* ldsE2   = smem + K_CODES * ESTR;        // K_CODES floats
    int*   ldsIdx  = (int*)(ldsE2 + K_CODES);      // TOK_PER_BLOCK ints
    float* ldsLoss = (float*)(ldsIdx + TOK_PER_BLOCK); // WAVES_PER_BLOCK floats

    const int tid  = threadIdx.x;
    const int lane = tid & 31;
    const int wave = tid >> 5;
    const int half = lane >> 4;   // 0: lanes 0-15, 1: lanes 16-31
    const int nn   = lane & 15;

    // ---- Stage codebook into LDS (padded rows), float4 coalesced ----
    const float4* emb4 = (const float4*)emb;
    for (int i = tid; i < K_CODES * C_DIM / 4; i += 256) {
        int code = i >> 4;          // 16 float4 per row
        int q4   = i & 15;
        float4 v = emb4[i];
        *(float4*)(ldsE + code * ESTR + q4 * 4) = v;
    }
    // ---- Per-code squared norms (from global, cached) ----
    for (int k = tid; k < K_CODES; k += 256) {
        const float* er = emb + k * C_DIM;
        float s = 0.f;
        #pragma unroll 8
        for (int c = 0; c < C_DIM; ++c) { float e = er[c]; s += e * e; }
        ldsE2[k] = s;
    }
    __syncthreads();

    // ---- Load this wave's 16-token A fragments (16x4 f32 WMMA layout) ----
    const int tokenBase = blockIdx.x * TOK_PER_BLOCK + wave * TOK_PER_WAVE;
    const int b  = tokenBase / N_SPATIAL;
    const int n0 = tokenBase % N_SPATIAL;
    const float* xb = x + (size_t)b * C_DIM * N_SPATIAL + n0;

    v2f A[16];
    #pragma unroll
    for (int kc = 0; kc < 16; ++kc) {
        int ch = kc * 4 + 2 * half;                 // VGPR0: K=0/2, VGPR1: K=1/3
        A[kc].x = xb[(size_t)ch * N_SPATIAL + nn];
        A[kc].y = xb[(size_t)(ch + 1) * N_SPATIAL + nn];
    }

    // ---- Fused distance GEMM + running argmin ----
    float minV[8];
    int   minI[8];
    #pragma unroll
    for (int r = 0; r < 8; ++r) { minV[r] = 3.4e38f; minI[r] = 0; }

    for (int kt = 0; kt < 64; ++kt) {
        v8f acc0 = {};
        v8f acc1 = {};
        const float* Bbase = ldsE + (kt * 16 + nn) * ESTR + 2 * half;
        #pragma unroll
        for (int kc = 0; kc < 16; kc += 2) {
            v2f B0 = *(const v2f*)(Bbase + kc * 4);
            v2f B1 = *(const v2f*)(Bbase + kc * 4 + 4);
            acc0 = __builtin_amdgcn_wmma_f32_16x16x4_f32(
                false, A[kc],     false, B0, (short)0, acc0, false, false);
            acc1 = __builtin_amdgcn_wmma_f32_16x16x4_f32(
                false, A[kc + 1], false, B1, (short)0, acc1, false, false);
        }
        float e2v = ldsE2[kt * 16 + nn];
        int   code = kt * 16 + nn;
        #pragma unroll
        for (int r = 0; r < 8; ++r) {
            float d = e2v - 2.0f * (acc0[r] + acc1[r]); // argmin-equivalent dist
            if (d < minV[r]) { minV[r] = d; minI[r] = code; }
        }
    }

    // ---- Cross-lane min+index reduce within each 16-lane half ----
    #pragma unroll
    for (int r = 0; r < 8; ++r) {
        float v = minV[r]; int i = minI[r];
        #pragma unroll
        for (int m = 8; m >= 1; m >>= 1) {
            float ov = __shfl_xor(v, m, 32);
            int   oi = __shfl_xor(i, m, 32);
            if (ov < v || (ov == v && oi < i)) { v = ov; i = oi; }
        }
        minV[r] = v; minI[r] = i;
    }
    // token for (row r, half h) = r + 8*h; publish to LDS
    if (nn == 0) {
        #pragma unroll
        for (int r = 0; r < 8; ++r)
            ldsIdx[wave * 16 + half * 8 + r] = minI[r];
    }
    __syncthreads();

    // ---- Gather codebook rows, write quant ([B,C,D,H,W]), accumulate loss ----
    const int myIdx = ldsIdx[wave * 16 + nn];   // token nn of this wave
    float* qb = outQ + (size_t)b * C_DIM * N_SPATIAL + n0;
    float lossAcc = 0.f;
    #pragma unroll 8
    for (int cc = half; cc < C_DIM; cc += 2) {  // half0: even ch, half1: odd ch
        float q  = ldsE[myIdx * ESTR + cc];
        float xv = xb[(size_t)cc * N_SPATIAL + nn];
        float dd = q - xv;
        lossAcc += dd * dd;
        qb[(size_t)cc * N_SPATIAL + nn] = q;
    }

    // indices output (as float), lanes 0-15 cover the wave's 16 tokens
    if (half == 0) outIdx[tokenBase + nn] = (float)myIdx;

    // ---- Deterministic loss reduction: wave -> block -> workspace ----
    #pragma unroll
    for (int m = 16; m >= 1; m >>= 1) lossAcc += __shfl_xor(lossAcc, m, 32);
    if (lane == 0) ldsLoss[wave] = lossAcc;
    __syncthreads();
    if (tid == 0) {
        float s = 0.f;
        #pragma unroll
        for (int w = 0; w < WAVES_PER_BLOCK; ++w) s += ldsLoss[w];
        wsPartial[blockIdx.x] = s;
    }
}

// ---------------------------------------------------------------------------
// Finalize: deterministic fixed-order sum of block partials -> mean losses
// ---------------------------------------------------------------------------
__global__ void vq_finalize(const float* __restrict__ wsPartial,
                            float* __restrict__ outLoss)
{
    __shared__ float sh[256];
    int tid = threadIdx.x;
    float s = 0.f;
    for (int i = tid; i < NBLOCKS; i += 256) s += wsPartial[i];
    sh[tid] = s;
    __syncthreads();
    for (int step = 128; step >= 1; step >>= 1) {
        if (tid < step) sh[tid] += sh[tid + step];
        __syncthreads();
    }
    if (tid == 0) {
        float mean = sh[0] / (float)QSIZE;  // mean over B*N*C = 16,777,216
        outLoss[0] = mean;                  // codebook_loss
        outLoss[1] = mean;                  // commit_loss (identical forward)
    }
}

extern "C" void kernel_launch(void* const* d_in, const int* in_sizes, int n_in,
                              void* d_out, int out_size, void* d_ws, size_t ws_size,
                              hipStream_t stream) {
    const float* x   = (const float*)d_in[0];   // [2,64,32,64,64] f32
    const float* emb = (const float*)d_in[1];   // [1024,64] f32
    float* out     = (float*)d_out;
    float* outQ    = out;                       // quant, QSIZE floats
    float* outLoss = out + QSIZE;               // 2 scalars
    float* outIdx  = out + QSIZE + 2;           // 262144 indices as float
    float* ws      = (float*)d_ws;              // NBLOCKS partial sums

    size_t smemBytes = (size_t)(K_CODES * ESTR + K_CODES) * sizeof(float)
                     + (size_t)TOK_PER_BLOCK * sizeof(int)
                     + (size_t)WAVES_PER_BLOCK * sizeof(float);   // ~283 KB < 320 KB

    vq_main<<<NBLOCKS, 256, smemBytes, stream>>>(x, emb, outQ, outIdx, ws);
    vq_finalize<<<1, 256, 0, stream>>>(ws, outLoss);
}